// HCI_2_add_77601469104875
// MI455X (gfx1250) — compile-verified
//
#include <hip/hip_runtime.h>

// ---------------------------------------------------------------------------
// Hierarchical 2-additive Choquet integral, 16-ary tree, depth 6, fp32.
// One kernel per level. Each wave handles 16 nodes:
//   - theta rows (16 x 136 f32) + children (16 x 16 f32) async-copied to LDS
//     via global_load_async_to_lds_b128 (ASYNCcnt path)
//   - softmax over 136 logits (2-pass, pair-lane split, shfl_xor(16) combine)
//   - 136-length weighted dot done as 34 chained V_WMMA_F32_16X16X4_F32 with
//     B = ones (exact row-sum on the matrix pipe, fp32 throughout)
// ---------------------------------------------------------------------------

typedef float v2f __attribute__((ext_vector_type(2)));
typedef float v4f __attribute__((ext_vector_type(4)));
typedef float v8f __attribute__((ext_vector_type(8)));

#define WAVES 4
#define THETA_PAD 140                       // dwords per theta row in LDS
#define CHILD_PAD 20                        // dwords per child row in LDS
#define WAVE_LDS (16 * THETA_PAD + 16 * CHILD_PAD)   // 2560 dwords / wave

// Static upper-triangle pair table (np.triu_indices(16, k=1) order).
struct PairTab {
  unsigned char pi[120];
  unsigned char pj[120];
  constexpr PairTab() : pi(), pj() {
    int p = 0;
    for (int a = 0; a < 16; ++a)
      for (int b = a + 1; b < 16; ++b) { pi[p] = (unsigned char)a; pj[p] = (unsigned char)b; ++p; }
  }
};
__device__ constexpr PairTab PT{};

// gfx1250 async global->LDS path (guarded so compile never hard-fails).
#if defined(__gfx1250__) && __has_builtin(__builtin_amdgcn_global_load_async_to_lds_b128) && \
    __has_builtin(__builtin_amdgcn_s_wait_asynccnt)
#define USE_ASYNC 1
// Parameter types per hipcc diagnostic: pointer-to int __attribute__((vector_size(16)))
// in address_space(1) for the global source, address_space(3) for the LDS dest.
typedef int v4i_gv __attribute__((vector_size(16)));
typedef __attribute__((address_space(1))) v4i_gv* gvec_ptr;
typedef __attribute__((address_space(3))) v4i_gv* lvec_ptr;
#else
#define USE_ASYNC 0
#endif

__device__ __forceinline__ void cp16(const float* gsrc, float* ldst) {
#if USE_ASYNC
  __builtin_amdgcn_global_load_async_to_lds_b128((gvec_ptr)gsrc, (lvec_ptr)ldst, 0, 0);
#else
  *(v4f*)ldst = *(const v4f*)gsrc;
#endif
}

// Feature k of a node given its 16 children (k folded to a constant by unroll).
__device__ __forceinline__ float featv(const float (&x)[16], int k) {
  if (k < 16) return x[k];
  const int p = k - 16;
  return fminf(x[PT.pi[p]], x[PT.pj[p]]);
}

__global__ __launch_bounds__(32 * WAVES) void choquet_level_kernel(
    const float* __restrict__ vin, const float* __restrict__ theta,
    float* __restrict__ vout, int n_nodes) {
  __shared__ float lds[WAVES * WAVE_LDS];

  const int lane = threadIdx.x & 31;
  const int wave = threadIdx.x >> 5;
  const int base = (blockIdx.x * WAVES + wave) * 16;   // first node of this wave
  if (base >= n_nodes) return;                          // wave-uniform exit

  float* shT = lds + wave * WAVE_LDS;
  float* shC = shT + 16 * THETA_PAD;

  const size_t thetaCount = (size_t)n_nodes * 136;
  const size_t childCount = (size_t)n_nodes * 16;

  // ---- stage theta: 16 rows x 136 f32 = 544 float4 chunks, 17 per lane ----
  {
    const size_t tb = (size_t)base * 136;
#pragma unroll
    for (int w = 0; w < 17; ++w) {
      const int q = lane + 32 * w;           // chunk id in [0,544)
      const int row = q / 34, c4 = q % 34;   // 136 = 34 float4 per row
      size_t gi = tb + (size_t)(4 * q);
      if (gi + 4 > thetaCount) gi = thetaCount - 4;   // in-bounds clamp (level 6)
      cp16(theta + gi, shT + row * THETA_PAD + c4 * 4);
    }
    // ---- stage children: 256 f32 = 64 float4 chunks, 2 per lane ----
    const size_t cb = (size_t)base * 16;
#pragma unroll
    for (int w = 0; w < 2; ++w) {
      const int q = lane + 32 * w;           // chunk id in [0,64)
      const int node = q >> 2, col = (q & 3) * 4;
      size_t gi = cb + (size_t)(4 * q);
      if (gi + 4 > childCount) gi = childCount - 4;
      cp16(vin + gi, shC + node * CHILD_PAD + col);
    }
  }
#if USE_ASYNC
  __builtin_amdgcn_s_wait_asynccnt(0);
#endif

  const int nl = lane & 15;    // node slot (matches WMMA A-matrix row M)
  const int half = lane >> 4;  // 0: k = 4t,4t+1   1: k = 4t+2,4t+3  (A K-layout)

  // Children of this lane's node -> registers (b128 LDS reads, 16B aligned).
  float x[16];
  {
    const float* cr = shC + nl * CHILD_PAD;
#pragma unroll
    for (int c = 0; c < 4; ++c) {
      const v4f v = *(const v4f*)(cr + 4 * c);
      x[4 * c + 0] = v.x; x[4 * c + 1] = v.y; x[4 * c + 2] = v.z; x[4 * c + 3] = v.w;
    }
  }

  const float* tr = shT + nl * THETA_PAD + 2 * half;

  // ---- pass A: row max of theta (softmax stabilization) ----
  float mx = -3.402823466e38f;
#pragma unroll
  for (int t = 0; t < 34; ++t) {
    const v2f th = *(const v2f*)(tr + 4 * t);
    mx = fmaxf(mx, fmaxf(th.x, th.y));
  }
  mx = fmaxf(mx, __shfl_xor(mx, 16, 32));

  // ---- pass B: exp-sum + weighted-feature products, row-summed by WMMA ----
  float S = 0.0f;
  v8f acc = {0.f, 0.f, 0.f, 0.f, 0.f, 0.f, 0.f, 0.f};
  const v2f bones = {1.0f, 1.0f};            // B = all-ones 4x16 (layout-free)
#pragma unroll
  for (int t = 0; t < 34; ++t) {
    const v2f th = *(const v2f*)(tr + 4 * t);
    const float e0 = __expf(th.x - mx);
    const float e1 = __expf(th.y - mx);
    S += e0 + e1;
    const int ka0 = 4 * t, ka1 = 4 * t + 1, kb0 = 4 * t + 2, kb1 = 4 * t + 3;
    const float f0 = half ? featv(x, kb0) : featv(x, ka0);
    const float f1 = half ? featv(x, kb1) : featv(x, ka1);
    v2f a; a.x = e0 * f0; a.y = e1 * f1;     // A[M=nl, K] per documented layout
    acc = __builtin_amdgcn_wmma_f32_16x16x4_f32(false, a, false, bones,
                                                (short)0, acc, false, false);
  }
  S += __shfl_xor(S, 16, 32);
  const float invS = 1.0f / S;

  // D VGPR r: lanes 0-15 hold M=r, lanes 16-31 hold M=8+r (columns identical).
  // Lane r stores node base+r; lane 24+r stores node base+8+r (its own S!).
#pragma unroll
  for (int r = 0; r < 8; ++r) {
    if (lane == r || lane == r + 24) {
      const int node = base + (lane & 15);
      if (node < n_nodes) vout[node] = acc[r] * invS;
    }
  }
}

// ---------------------------------------------------------------------------
// Host side: 6 chained level launches, ping-pong through workspace.
// Scratch use: bufA = 4 MB (level-1 out), bufB = 256 KB (level-2 out).
// ---------------------------------------------------------------------------
static inline void run_level(const float* in, const float* th, float* out,
                             int n_nodes, hipStream_t stream) {
  const int groups = (n_nodes + 15) / 16;
  const int blocks = (groups + WAVES - 1) / WAVES;
  choquet_level_kernel<<<blocks, 32 * WAVES, 0, stream>>>(in, th, out, n_nodes);
}

extern "C" void kernel_launch(void* const* d_in, const int* in_sizes, int n_in,
                              void* d_out, int out_size, void* d_ws, size_t ws_size,
                              hipStream_t stream) {
  (void)in_sizes; (void)n_in; (void)out_size; (void)ws_size;
  const float* x   = (const float*)d_in[0];
  const float* th1 = (const float*)d_in[1];
  const float* th2 = (const float*)d_in[2];
  const float* th3 = (const float*)d_in[3];
  const float* th4 = (const float*)d_in[4];
  const float* th5 = (const float*)d_in[5];
  const float* th6 = (const float*)d_in[6];

  float* bufA = (float*)d_ws;
  float* bufB = bufA + (1u << 20);   // after 1,048,576 floats (4 MB)

  run_level(x,    th1, bufA, 1048576, stream);  // 16^5 nodes
  run_level(bufA, th2, bufB, 65536,   stream);  // 16^4
  run_level(bufB, th3, bufA, 4096,    stream);  // 16^3
  run_level(bufA, th4, bufB, 256,     stream);  // 16^2
  run_level(bufB, th5, bufA, 16,      stream);  // 16^1
  run_level(bufA, th6, (float*)d_out, 1, stream);  // root
}